// MTvae_73461120630899
// MI455X (gfx1250) — compile-verified
//
#include <hip/hip_runtime.h>
#include <hip/hip_bf16.h>
#include <math.h>

// Problem constants (from reference)
#define K_CL 16
#define DZ   64
#define DT   1024
#define NT   2048
#define NI   4096
#define MU_P_WT 5.0f
#define COS_EPS 1e-6f

typedef float v2f __attribute__((ext_vector_type(2)));
typedef float v8f __attribute__((ext_vector_type(8)));
typedef unsigned int u32x4 __attribute__((ext_vector_type(4)));
typedef int i32x4 __attribute__((ext_vector_type(4)));
typedef int i32x8 __attribute__((ext_vector_type(8)));

// D = A(16x4 f32) * B(4x16 f32) + C(16x16 f32), full-precision WMMA.
__device__ __forceinline__ v8f wmma_f32_k4(v2f a, v2f b, v8f c) {
  return __builtin_amdgcn_wmma_f32_16x16x4_f32(
      /*neg_a=*/false, a, /*neg_b=*/false, b,
      /*c_mod=*/(short)0, c, /*reuse_a=*/false, /*reuse_b=*/false);
}

// ---------------------------------------------------------------------------
// Tensor Data Mover: async 2-D f32 tile load, global -> LDS.
// D# packing per CDNA5 ISA 8.3/8.4: group0 = {count=1, lds_addr,
// global_addr[56:0], type=2}; group1 = {data_size=2 (4B), tensor_dim0/1,
// tile_dim0/1, tensor_dim0_stride}; groups 2/3 zero (2-D tile).
// Rows (dim1) x row-length (dim0) stored contiguously at lds_off.
// ---------------------------------------------------------------------------
__device__ __forceinline__ void tdm_load_tile_f32(
    const float* gtile, unsigned lds_off, unsigned tile_d0, unsigned tile_d1,
    unsigned tensor_d0, unsigned tensor_d1, unsigned stride0_elems) {
  const unsigned long long ga = (unsigned long long)(size_t)gtile;
  u32x4 g0;
  g0.x = 1u;                                          // count=1 (user D#)
  g0.y = lds_off;                                     // lds_addr (bytes)
  g0.z = (unsigned)(ga & 0xFFFFFFFFull);              // global_addr[31:0]
  g0.w = (unsigned)((ga >> 32) & 0x1FFFFFFull)        // global_addr[56:32]
         | (2u << 30);                                // type=2 ("image")
  i32x8 g1;
  g1[0] = (int)(2u << 16);                            // data_size = 4 bytes
  g1[1] = (int)((tensor_d0 & 0xFFFFu) << 16);         // tensor_dim0[15:0]
  g1[2] = (int)((tensor_d0 >> 16) | ((tensor_d1 & 0xFFFFu) << 16));
  g1[3] = (int)((tensor_d1 >> 16) | (tile_d0 << 16)); // tile_dim0
  g1[4] = (int)(tile_d1 & 0xFFFFu);                   // tile_dim1 (tile_dim2=0)
  g1[5] = (int)stride0_elems;                         // tensor_dim0_stride lo
  g1[6] = 0;
  g1[7] = 0;
  i32x4 z4 = {0, 0, 0, 0};
#if defined(__clang_major__) && (__clang_major__ >= 23)
  i32x8 z8 = {0, 0, 0, 0, 0, 0, 0, 0};
  __builtin_amdgcn_tensor_load_to_lds(g0, g1, z4, z4, z8, 0);
#else
  __builtin_amdgcn_tensor_load_to_lds(g0, g1, z4, z4, 0);
#endif
}

// Low 32 bits of a generic shared-memory pointer == LDS byte offset.
__device__ __forceinline__ unsigned lds_off_of(const void* p) {
  return (unsigned)(size_t)p;
}

// ---------------------------------------------------------------------------
// encode_t: mu_zt / logvar_zt = input_treat @ W.T + b   ([2048,1024]x[1024,64])
// One wave per 16x16 output tile; both matrices share the A loads.
// ---------------------------------------------------------------------------
__global__ __launch_bounds__(32) void encode_t_kernel(
    const float* __restrict__ treat, const float* __restrict__ Wmu,
    const float* __restrict__ bmu,   const float* __restrict__ Wlv,
    const float* __restrict__ blv,   float* __restrict__ mu_zt,
    float* __restrict__ lv_zt) {
  const int lane = threadIdx.x & 31;
  const int m0 = blockIdx.x * 16;          // NT tile
  const int n0 = blockIdx.y * 16;          // DZ tile
  const int mn = lane & 15;
  const int koff = (lane >> 4) << 1;       // 0 or 2
  const int mhi = (lane >> 4) << 3;        // 0 or 8
  v8f accm = {}, accl = {};
  const float* arow = treat + (size_t)(m0 + mn) * DT + koff;
  const float* bmr  = Wmu   + (size_t)(n0 + mn) * DT + koff;   // B[k][n]=W[n][k]
  const float* blr  = Wlv   + (size_t)(n0 + mn) * DT + koff;
  for (int kk = 0; kk < DT; kk += 4) {
    v2f a  = *(const v2f*)(arow + kk);
    v2f bm = *(const v2f*)(bmr + kk);
    v2f bl = *(const v2f*)(blr + kk);
    accm = wmma_f32_k4(a, bm, accm);
    accl = wmma_f32_k4(a, bl, accl);
  }
  const float biasM = bmu[n0 + mn], biasL = blv[n0 + mn];
  for (int v = 0; v < 8; ++v) {
    const int m = v + mhi;
    mu_zt[(size_t)(m0 + m) * DZ + n0 + mn] = accm[v] + biasM;
    lv_zt[(size_t)(m0 + m) * DZ + n0 + mn] = accl[v] + biasL;
  }
}

// ---------------------------------------------------------------------------
// qc: cosine-sim vs 5*mu_p, *10, softmax -> cates, softmax again -> c_sample
// ---------------------------------------------------------------------------
__global__ __launch_bounds__(64) void qc_kernel(
    const float* __restrict__ mu_zt, const float* __restrict__ mu_p,
    float* __restrict__ cates, float* __restrict__ csamp) {
  const int t = blockIdx.x * 64 + threadIdx.x;
  if (t >= NT) return;
  const float* z = mu_zt + (size_t)t * DZ;
  float zr[DZ], nz = 0.f;
  for (int d = 0; d < DZ; ++d) { zr[d] = z[d]; nz += zr[d] * zr[d]; }
  nz = sqrtf(nz);
  float qc[K_CL];
  for (int k = 0; k < K_CL; ++k) {
    const float* c = mu_p + k * DZ;
    float num = 0.f, nc = 0.f;
    for (int d = 0; d < DZ; ++d) {
      const float cv = MU_P_WT * c[d];
      num += zr[d] * cv;
      nc += cv * cv;
    }
    qc[k] = 10.f * num / fmaxf(nz * sqrtf(nc), COS_EPS);
  }
  float mx = qc[0];
  for (int k = 1; k < K_CL; ++k) mx = fmaxf(mx, qc[k]);
  float s = 0.f;
  for (int k = 0; k < K_CL; ++k) { qc[k] = __expf(qc[k] - mx); s += qc[k]; }
  const float inv = 1.f / s;
  float mx2 = -1e30f;
  for (int k = 0; k < K_CL; ++k) {
    qc[k] *= inv;
    cates[(size_t)t * K_CL + k] = qc[k];
    mx2 = fmaxf(mx2, qc[k]);
  }
  float s2 = 0.f;
  for (int k = 0; k < K_CL; ++k) { qc[k] = __expf(qc[k] - mx2); s2 += qc[k]; }
  const float inv2 = 1.f / s2;
  for (int k = 0; k < K_CL; ++k) csamp[(size_t)t * K_CL + k] = qc[k] * inv2;
}

// ---------------------------------------------------------------------------
// encode_i: mu_zi[k]/logvar_zi[k] = (ins * cs_k) @ W.T + b, cs folded into B.
// Grid (NI/16, K). Block = 4 waves = 4 DZ tiles sharing one 16-row stripe of
// input_ins, staged in LDS by the TDM (double-buffered, TENSORcnt-paced).
// ---------------------------------------------------------------------------
#define TCH 512                       // t-chunk staged per TDM descriptor
#define NCH (NT / TCH)

__global__ __launch_bounds__(128) void encode_i_kernel(
    const float* __restrict__ ins, const float* __restrict__ Wmu,
    const float* __restrict__ bmu, const float* __restrict__ Wlv,
    const float* __restrict__ blv, const float* __restrict__ csamp,
    float* __restrict__ mu_zi, float* __restrict__ lv_zi,
    float* __restrict__ zi) {
  __shared__ float shA[2][16 * TCH];  // 64 KB double buffer
  const int lane = threadIdx.x & 31;
  const int wav  = threadIdx.x >> 5;
  const int m0 = blockIdx.x * 16;          // NI tile
  const int kc = blockIdx.y;               // cluster
  const int n0 = wav * 16;                 // DZ tile (one per wave)
  const int mn = lane & 15;
  const int koff = (lane >> 4) << 1;
  const int mhi = (lane >> 4) << 3;
  const bool w0 = (threadIdx.x < 32);      // wave 0 drives the TDM

  v8f accm = {}, accl = {};
  const float* wmr = Wmu + (size_t)(n0 + mn) * NT + koff;
  const float* wlr = Wlv + (size_t)(n0 + mn) * NT + koff;
  const float* cs  = csamp + (size_t)koff * K_CL + kc;

  if (w0)
    tdm_load_tile_f32(ins + (size_t)m0 * NT, lds_off_of(&shA[0][0]),
                      TCH, 16, NT, NI, NT);
  for (int c = 0; c < NCH; ++c) {
    if (w0) {
      if (c + 1 < NCH) {
        tdm_load_tile_f32(ins + (size_t)m0 * NT + (size_t)(c + 1) * TCH,
                          lds_off_of(&shA[(c + 1) & 1][0]), TCH, 16, NT, NI, NT);
        __builtin_amdgcn_s_wait_tensorcnt(1);   // chunk c landed
      } else {
        __builtin_amdgcn_s_wait_tensorcnt(0);
      }
    }
    __syncthreads();                            // shA[c&1] visible to all waves
    const float* sa = &shA[c & 1][(size_t)mn * TCH + koff];
    const int t0 = c * TCH;
    for (int kkl = 0; kkl < TCH; kkl += 4) {
      v2f a = *(const v2f*)(sa + kkl);          // LDS A fragment
      const int kk = t0 + kkl;
      const float c0 = cs[(size_t)kk * K_CL];
      const float c1 = cs[(size_t)(kk + 1) * K_CL];
      v2f wm = *(const v2f*)(wmr + kk);
      v2f wl = *(const v2f*)(wlr + kk);
      v2f bmv;  bmv.x  = wm.x * c0; bmv.y  = wm.y * c1;
      v2f blv2; blv2.x = wl.x * c0; blv2.y = wl.y * c1;
      accm = wmma_f32_k4(a, bmv, accm);
      accl = wmma_f32_k4(a, blv2, accl);
    }
    __syncthreads();                            // done reading before reuse
  }
  const float biasM = bmu[n0 + mn], biasL = blv[n0 + mn];
  const size_t base = (size_t)kc * NI * DZ;
  for (int v = 0; v < 8; ++v) {
    const int m = v + mhi;
    const size_t idx = base + (size_t)(m0 + m) * DZ + n0 + mn;
    const float mv = accm[v] + biasM;
    mu_zi[idx] = mv;
    zi[idx] = mv;
    lv_zi[idx] = accl[v] + biasL;
  }
}

// ---------------------------------------------------------------------------
// decoder: a_pred[n,t] = sum_k sigmoid(zi_k[n,:]·zt[t,:]) * c_sample[t,k]
// 8 waves/block share the 16x64 zi tile per cluster k (TDM double-buffered in
// LDS). zt fragment + c_sample row are register-resident across the k loop.
// Logits tensor never materialized.
// ---------------------------------------------------------------------------
__global__ __launch_bounds__(256) void decoder_kernel(
    const float* __restrict__ zi, const float* __restrict__ zt,
    const float* __restrict__ csamp, float* __restrict__ a_pred) {
  __shared__ float shZ[2][16 * DZ];   // 8 KB double buffer
  const int wave = threadIdx.x >> 5;
  const int lane = threadIdx.x & 31;
  const int m0 = blockIdx.x * 16;                    // NI tile (shared)
  const int n0 = (blockIdx.y * 8 + wave) * 16;       // NT tile (per wave)
  const int mn = lane & 15;
  const int koff = (lane >> 4) << 1;
  const int mhi = (lane >> 4) << 3;
  const bool w0 = (threadIdx.x < 32);

  v2f ztr[16];                                       // zt fragment, 32 VGPRs
  {
    const float* zr = zt + (size_t)(n0 + mn) * DZ + koff;
    for (int kk = 0; kk < 16; ++kk) ztr[kk] = *(const v2f*)(zr + kk * 4);
  }
  float csr[K_CL];                                   // per-column k weights
  {
    const float* cr = csamp + (size_t)(n0 + mn) * K_CL;
    for (int k = 0; k < K_CL; ++k) csr[k] = cr[k];
  }
  float ap[8] = {0.f, 0.f, 0.f, 0.f, 0.f, 0.f, 0.f, 0.f};

  if (w0)
    tdm_load_tile_f32(zi + (size_t)m0 * DZ, lds_off_of(&shZ[0][0]),
                      DZ, 16, DZ, K_CL * NI, DZ);
  for (int kc = 0; kc < K_CL; ++kc) {
    if (w0) {
      if (kc + 1 < K_CL) {
        tdm_load_tile_f32(zi + ((size_t)(kc + 1) * NI + m0) * DZ,
                          lds_off_of(&shZ[(kc + 1) & 1][0]),
                          DZ, 16, DZ, K_CL * NI, DZ);
        __builtin_amdgcn_s_wait_tensorcnt(1);
      } else {
        __builtin_amdgcn_s_wait_tensorcnt(0);
      }
    }
    __syncthreads();
    const float* sa = &shZ[kc & 1][(size_t)mn * DZ + koff];
    v8f acc = {};
    for (int kk = 0; kk < DZ; kk += 4)
      acc = wmma_f32_k4(*(const v2f*)(sa + kk), ztr[kk >> 2], acc);
    const float cw = csr[kc];
    for (int v = 0; v < 8; ++v)
      ap[v] += cw * (1.0f / (1.0f + __expf(-acc[v])));
    __syncthreads();
  }
  for (int v = 0; v < 8; ++v)
    a_pred[(size_t)(m0 + v + mhi) * NT + n0 + mn] = ap[v];
}

// ---------------------------------------------------------------------------
// a_reconst = sigmoid(zt @ W_arec.T + b)   [2048,64]x[64,1024]
// ---------------------------------------------------------------------------
__global__ __launch_bounds__(32) void arec_kernel(
    const float* __restrict__ zt, const float* __restrict__ Wa,
    const float* __restrict__ ba, float* __restrict__ out) {
  const int lane = threadIdx.x & 31;
  const int m0 = blockIdx.x * 16;          // NT tile
  const int n0 = blockIdx.y * 16;          // DT tile
  const int mn = lane & 15;
  const int koff = (lane >> 4) << 1;
  const int mhi = (lane >> 4) << 3;
  v8f acc = {};
  const float* arow = zt + (size_t)(m0 + mn) * DZ + koff;
  const float* brow = Wa + (size_t)(n0 + mn) * DZ + koff;
  for (int kk = 0; kk < DZ; kk += 4) {
    v2f a = *(const v2f*)(arow + kk);
    v2f b = *(const v2f*)(brow + kk);
    acc = wmma_f32_k4(a, b, acc);
  }
  const float bias = ba[n0 + mn];
  for (int v = 0; v < 8; ++v) {
    const float x = acc[v] + bias;
    out[(size_t)(m0 + v + mhi) * DT + n0 + mn] = 1.0f / (1.0f + __expf(-x));
  }
}

// ---------------------------------------------------------------------------
// a_zt = input_ins @ zt   [4096,2048]x[2048,64]
// ---------------------------------------------------------------------------
__global__ __launch_bounds__(32) void azt_kernel(
    const float* __restrict__ ins, const float* __restrict__ zt,
    float* __restrict__ a_zt) {
  const int lane = threadIdx.x & 31;
  const int m0 = blockIdx.x * 16;          // NI tile
  const int n0 = blockIdx.y * 16;          // DZ tile
  const int mn = lane & 15;
  const int koff = (lane >> 4) << 1;
  const int mhi = (lane >> 4) << 3;
  v8f acc = {};
  const float* arow = ins + (size_t)(m0 + mn) * NT + koff;
  const float* bcol = zt + (size_t)koff * DZ + n0 + mn;   // B[t][d]=zt[t*DZ+d]
  for (int kk = 0; kk < NT; kk += 4) {
    v2f a = *(const v2f*)(arow + kk);
    v2f b;
    b.x = bcol[(size_t)kk * DZ];
    b.y = bcol[(size_t)(kk + 1) * DZ];
    acc = wmma_f32_k4(a, b, acc);
  }
  for (int v = 0; v < 8; ++v)
    a_zt[(size_t)(m0 + v + mhi) * DZ + n0 + mn] = acc[v];
}

// ---------------------------------------------------------------------------
// rep_w1 = a_zt @ W_y1.T + b_y1   [4096,64]x[64,1024]
// ---------------------------------------------------------------------------
__global__ __launch_bounds__(32) void repw1_kernel(
    const float* __restrict__ a_zt, const float* __restrict__ Wy1,
    const float* __restrict__ by1, float* __restrict__ rep) {
  const int lane = threadIdx.x & 31;
  const int m0 = blockIdx.x * 16;          // NI tile
  const int n0 = blockIdx.y * 16;          // K*DZ tile
  const int mn = lane & 15;
  const int koff = (lane >> 4) << 1;
  const int mhi = (lane >> 4) << 3;
  v8f acc = {};
  const float* arow = a_zt + (size_t)(m0 + mn) * DZ + koff;
  const float* brow = Wy1 + (size_t)(n0 + mn) * DZ + koff;
  for (int kk = 0; kk < DZ; kk += 4) {
    v2f a = *(const v2f*)(arow + kk);
    v2f b = *(const v2f*)(brow + kk);
    acc = wmma_f32_k4(a, b, acc);
  }
  const float bias = by1[n0 + mn];
  for (int v = 0; v < 8; ++v)
    rep[(size_t)(m0 + v + mhi) * (K_CL * DZ) + n0 + mn] = acc[v] + bias;
}

// ---------------------------------------------------------------------------
// mu_y[n] = sum_j zi_all[n,j]*(rep_w1[n,j] + W_y2[j]) + b_y2 ; logvar_y = 1
// zi_all[n, k*DZ+d] = mu_zi[k,n,d]
// ---------------------------------------------------------------------------
__global__ __launch_bounds__(256) void ymu_kernel(
    const float* __restrict__ rep, const float* __restrict__ mu_zi,
    const float* __restrict__ Wy2, const float* __restrict__ by2,
    float* __restrict__ mu_y, float* __restrict__ lv_y) {
  const int n = blockIdx.x;
  __shared__ float red[256];
  float s = 0.f;
  for (int j = threadIdx.x; j < K_CL * DZ; j += 256) {
    const int k = j >> 6, d = j & 63;
    const float z = mu_zi[((size_t)k * NI + n) * DZ + d];
    s += z * (rep[(size_t)n * (K_CL * DZ) + j] + Wy2[j]);
  }
  red[threadIdx.x] = s;
  __syncthreads();
  for (int off = 128; off > 0; off >>= 1) {
    if (threadIdx.x < off) red[threadIdx.x] += red[threadIdx.x + off];
    __syncthreads();
  }
  if (threadIdx.x == 0) {
    mu_y[n] = red[0] + by2[0];
    lv_y[n] = 1.0f;
  }
}

// Pass-through copies of the prior parameters.
__global__ __launch_bounds__(256) void copy_kernel(
    const float* __restrict__ a, const float* __restrict__ b,
    float* __restrict__ oa, float* __restrict__ ob, int n) {
  const int i = blockIdx.x * 256 + threadIdx.x;
  if (i < n) { oa[i] = a[i]; ob[i] = b[i]; }
}

extern "C" void kernel_launch(void* const* d_in, const int* in_sizes, int n_in,
                              void* d_out, int out_size, void* d_ws, size_t ws_size,
                              hipStream_t stream) {
  const float* input_ins    = (const float*)d_in[0];
  const float* input_treat  = (const float*)d_in[1];
  const float* W_mu_zt      = (const float*)d_in[2];
  const float* b_mu_zt      = (const float*)d_in[3];
  const float* W_lv_zt      = (const float*)d_in[4];
  const float* b_lv_zt      = (const float*)d_in[5];
  const float* W_arec       = (const float*)d_in[6];
  const float* b_arec       = (const float*)d_in[7];
  const float* W_mu_zi      = (const float*)d_in[8];
  const float* b_mu_zi      = (const float*)d_in[9];
  const float* W_lv_zi      = (const float*)d_in[10];
  const float* b_lv_zi      = (const float*)d_in[11];
  const float* mu_p_zt      = (const float*)d_in[12];
  const float* logvar_p_zt  = (const float*)d_in[13];
  const float* W_y1         = (const float*)d_in[14];
  const float* b_y1         = (const float*)d_in[15];
  const float* W_y2         = (const float*)d_in[16];
  const float* b_y2         = (const float*)d_in[17];

  float* out = (float*)d_out;
  float* o_mu_zt  = out;              // [2048,64]
  float* o_lv_zt  = out + 131072;     // [2048,64]
  float* o_mu_p   = out + 262144;     // [16,64]
  float* o_lv_p   = out + 263168;     // [16,64]
  float* o_cates  = out + 264192;     // [2048,16]
  float* o_mu_zi  = out + 296960;     // [16,4096,64]
  float* o_lv_zi  = out + 4491264;    // [16,4096,64]
  float* o_zi     = out + 8685568;    // [16,4096,64]
  float* o_a_pred = out + 12879872;   // [4096,2048]
  float* o_mu_y   = out + 21268480;   // [4096,1]
  float* o_lv_y   = out + 21272576;   // [4096,1]
  float* o_arec   = out + 21276672;   // [2048,1024]

  float* ws = (float*)d_ws;
  float* w_csample = ws;                       // 32768   floats
  float* w_azt     = ws + 32768;               // 262144  floats
  float* w_rep     = ws + 32768 + 262144;      // 4194304 floats

  encode_t_kernel<<<dim3(NT / 16, DZ / 16), 32, 0, stream>>>(
      input_treat, W_mu_zt, b_mu_zt, W_lv_zt, b_lv_zt, o_mu_zt, o_lv_zt);
  copy_kernel<<<dim3(4), 256, 0, stream>>>(mu_p_zt, logvar_p_zt, o_mu_p, o_lv_p,
                                           K_CL * DZ);
  qc_kernel<<<dim3(NT / 64), 64, 0, stream>>>(o_mu_zt, mu_p_zt, o_cates,
                                              w_csample);
  encode_i_kernel<<<dim3(NI / 16, K_CL), 128, 0, stream>>>(
      input_ins, W_mu_zi, b_mu_zi, W_lv_zi, b_lv_zi, w_csample, o_mu_zi,
      o_lv_zi, o_zi);
  decoder_kernel<<<dim3(NI / 16, NT / 128), 256, 0, stream>>>(
      o_mu_zi, o_mu_zt, w_csample, o_a_pred);
  arec_kernel<<<dim3(NT / 16, DT / 16), 32, 0, stream>>>(o_mu_zt, W_arec,
                                                         b_arec, o_arec);
  azt_kernel<<<dim3(NI / 16, DZ / 16), 32, 0, stream>>>(input_ins, o_mu_zt,
                                                        w_azt);
  repw1_kernel<<<dim3(NI / 16, (K_CL * DZ) / 16), 32, 0, stream>>>(
      w_azt, W_y1, b_y1, w_rep);
  ymu_kernel<<<dim3(NI), 256, 0, stream>>>(w_rep, o_mu_zi, W_y2, b_y2, o_mu_y,
                                           o_lv_y);
}